// SNN_37151467110513
// MI455X (gfx1250) — compile-verified
//
#include <hip/hip_runtime.h>

// ---------------------------------------------------------------------------
// SNN reference: 25 x 32768 sequential LIF steps, 2 layers (36->5->3).
// Plan:
//   Kernel 1 (WMMA, parallel): U[h][r] = (data @ w1^T + b1)[r][h], r<32768, h<5
//       using V_WMMA_F32_16X16X4_F32 (K=36 -> 9 k-steps). U stored transposed
//       (per-neuron contiguous) in d_ws for streaming float4 loads.
//   Kernel 2 (block-parallel scan): timeline 819200 steps cut into 1600 blocks
//       of 512 steps, each warm-started 256 steps early (beta=0.9 contraction:
//       0.9^256 ~ 2e-12 << f32 eps). One wave per block: lanes 0-4 run the 5
//       independent m1 chains, ballot -> 5-bit spike pattern (wave-uniform),
//       v_readlane pulls LUT[pattern] = w2@s1+b2 from lane registers, all
//       lanes update m2 redundantly, lanes 0-2 emit spikes (both tuple copies).
// ---------------------------------------------------------------------------

#define NUM_INPUTS   36
#define NUM_HIDDEN   5
#define NUM_OUTPUTS  3
#define ROWS         32768
#define NUM_STEPS    25
#define TOTAL_T      (ROWS * NUM_STEPS)      // 819200
#define BLK_L        512                     // emitted steps per block
#define WARM         256                     // warm-up steps (contraction)
#define NBLK         (TOTAL_T / BLK_L)       // 1600
#define BETA         0.9f
#define THRESH       1.0f

typedef __attribute__((ext_vector_type(2))) float v2f;
typedef __attribute__((ext_vector_type(8))) float v8f;

// ---------------------------------------------------------------------------
// Kernel 1: U = data @ w1^T + b1 via f32 WMMA. One wave per 16-row tile.
//   A tile: 16x36 f32 from data; B: 36x16 (w1^T zero-padded to N=16).
//   9 x V_WMMA_F32_16X16X4_F32 accumulate into 16x16 f32 C (cols 5..15 junk).
// ---------------------------------------------------------------------------
__global__ __launch_bounds__(32)
void snn_u_gemm(const float* __restrict__ data,   // [ROWS][36]
                const float* __restrict__ w1,     // [5][36]
                const float* __restrict__ b1,     // [5]
                float* __restrict__ U)            // [5][ROWS] transposed
{
    const int lane = threadIdx.x;        // 0..31
    const int tile = blockIdx.x;         // 0..ROWS/16-1
    const int M    = lane & 15;          // A row / C col index component
    const int half = lane >> 4;          // 0 or 1
    const int k0   = half * 2;           // K sub-offset within a k-step

    const float* arow = data + (tile * 16 + M) * NUM_INPUTS;

    // B operand: lane -> column n of w1^T (n = M); branchless pad for n>=5.
    const int   n     = M;
    const int   nc    = (n < NUM_HIDDEN) ? n : 0;
    const float nmask = (n < NUM_HIDDEN) ? 1.0f : 0.0f;
    const float* brow = w1 + nc * NUM_INPUTS;

    v8f acc = {};
#pragma unroll
    for (int kk = 0; kk < NUM_INPUTS / 4; ++kk) {   // 9 k-steps of K=4
        v2f a, b;
        a.x = arow[kk * 4 + k0 + 0];
        a.y = arow[kk * 4 + k0 + 1];
        b.x = nmask * brow[kk * 4 + k0 + 0];
        b.y = nmask * brow[kk * 4 + k0 + 1];
        // D = A x B + C   (EXEC all-1s: no divergent control flow here)
        acc = __builtin_amdgcn_wmma_f32_16x16x4_f32(
                  /*neg_a=*/false, a, /*neg_b=*/false, b,
                  /*c_mod=*/(short)0, acc, /*reuse_a=*/false, /*reuse_b=*/false);
    }

    const float bias = nmask * b1[nc];
    if (n < NUM_HIDDEN) {
#pragma unroll
        for (int v = 0; v < 8; ++v) {
            const int m = v + half * 8;                  // C row = data row
            U[n * ROWS + tile * 16 + m] = acc[v] + bias; // transposed store
        }
    }
}

// ---------------------------------------------------------------------------
// Kernel 2: block-parallel LIF scan. One wave (32 threads) per time block.
// ---------------------------------------------------------------------------
__global__ __launch_bounds__(32)
void snn_scan(const float* __restrict__ U,    // [5][ROWS]
              const float* __restrict__ w2,   // [3][5]
              const float* __restrict__ b2,   // [3]
              float* __restrict__ out)        // [2][TOTAL_T][3]
{
    const int lane = threadIdx.x;     // 0..31
    const int blk  = blockIdx.x;      // 0..NBLK-1

    // LUT in lane registers: lane p holds (w2 @ bits(p) + b2) for pattern p.
    float lx = b2[0], ly = b2[1], lz = b2[2];
#pragma unroll
    for (int h = 0; h < NUM_HIDDEN; ++h) {
        const float on = ((lane >> h) & 1) ? 1.0f : 0.0f;
        lx += on * w2[0 * NUM_HIDDEN + h];
        ly += on * w2[1 * NUM_HIDDEN + h];
        lz += on * w2[2 * NUM_HIDDEN + h];
    }

    // Lanes 0..4 each own one m1 neuron; others alias lane 0's stream (masked).
    const int   h      = (lane < NUM_HIDDEN) ? lane : 0;
    const float hvalid = (lane < NUM_HIDDEN) ? 1.0f : 0.0f;
    const float* Uh    = U + h * ROWS;

    const int emit0  = blk * BLK_L;
    const int tstart = emit0 - WARM;
    const int t0     = tstart < 0 ? 0 : tstart;   // multiple of 256 (or 0)
    const int tend   = emit0 + BLK_L;

    float m1 = 0.0f;
    float m2x = 0.0f, m2y = 0.0f, m2z = 0.0f;

    for (int t = t0; t < tend; t += 4) {
        const int idx = t & (ROWS - 1);           // 4-aligned, never wraps mid-chunk
        const float4 u4 = *reinterpret_cast<const float4*>(Uh + idx);
        const bool emit = (t >= emit0);           // block-aligned: uniform per chunk

#pragma unroll
        for (int j = 0; j < 4; ++j) {
            const float u = (j == 0) ? u4.x : (j == 1) ? u4.y : (j == 2) ? u4.z : u4.w;

            // ---- layer 1 (lanes 0..4 meaningful) ----
            m1 = fmaf(BETA, m1, u);
            const bool sp1 = (hvalid != 0.0f) && (m1 > THRESH);
            if (sp1) m1 -= THRESH;

            // wave-uniform 5-bit spike pattern
            const unsigned pat = __builtin_amdgcn_ballot_w32(sp1) & 31u;

            // LUT gather via v_readlane (uniform lane select -> SGPR)
            const float vx = __uint_as_float(
                __builtin_amdgcn_readlane(__float_as_uint(lx), pat));
            const float vy = __uint_as_float(
                __builtin_amdgcn_readlane(__float_as_uint(ly), pat));
            const float vz = __uint_as_float(
                __builtin_amdgcn_readlane(__float_as_uint(lz), pat));

            // ---- layer 2 (redundant across lanes; 3 scalar chains) ----
            m2x = fmaf(BETA, m2x, vx);
            m2y = fmaf(BETA, m2y, vy);
            m2z = fmaf(BETA, m2z, vz);
            const float sx = (m2x > THRESH) ? 1.0f : 0.0f;
            const float sy = (m2y > THRESH) ? 1.0f : 0.0f;
            const float sz = (m2z > THRESH) ? 1.0f : 0.0f;
            m2x -= sx; m2y -= sy; m2z -= sz;

            // ---- emit: lanes 0..2 write one component each, both copies ----
            if (emit && lane < NUM_OUTPUTS) {
                const float sv = (lane == 0) ? sx : (lane == 1) ? sy : sz;
                const int off = (t + j) * NUM_OUTPUTS + lane;
                out[off] = sv;
                out[TOTAL_T * NUM_OUTPUTS + off] = sv;
            }
        }
    }
}

extern "C" void kernel_launch(void* const* d_in, const int* in_sizes, int n_in,
                              void* d_out, int out_size, void* d_ws, size_t ws_size,
                              hipStream_t stream)
{
    (void)in_sizes; (void)n_in; (void)out_size; (void)ws_size;

    const float* data = (const float*)d_in[0];   // [32768][36]
    const float* w1   = (const float*)d_in[1];   // [5][36]
    const float* b1   = (const float*)d_in[2];   // [5]
    const float* w2   = (const float*)d_in[3];   // [3][5]
    const float* b2   = (const float*)d_in[4];   // [3]
    float* out = (float*)d_out;                  // 2 x 819200 x 3
    float* U   = (float*)d_ws;                   // 5 x 32768 f32 = 640 KB

    // Kernel 1: WMMA GEMM, one wave per 16-row tile.
    snn_u_gemm<<<ROWS / 16, 32, 0, stream>>>(data, w1, b1, U);

    // Kernel 2: 1600 independent warm-started scan blocks, one wave each.
    snn_scan<<<NBLK, 32, 0, stream>>>(U, w2, b2, out);
}